// LOT_60911226192408
// MI455X (gfx1250) — compile-verified
//
#include <hip/hip_runtime.h>
#include <math.h>

// ---------------- CDNA5 fp32 WMMA types ----------------
typedef __attribute__((ext_vector_type(2))) float v2f;
typedef __attribute__((ext_vector_type(8))) float v8f;

#define N_PTS 2048
#define M_PTS 2048
#define DIM   128
#define KA    64
#define KB    64
#define T_NITER 30
#define T_NINNER 100
#define TOLRATIO 1e-3f
#define CLIPV 1e8f
#define WPB 4   // waves per wmma block (128 threads)

// ---------------- workspace layout (float indices) ----------------
#define OFF_CX    0u
#define OFF_CY    8192u
#define OFF_SN    16384u
#define OFF_TN    18432u
#define OFF_CXN   20480u
#define OFF_CYN   20544u
#define OFF_KX    20608u     // 2048x64
#define OFF_KZ    151680u    // 64x64
#define OFF_KY    155776u    // 64x2048
#define OFF_AX    286848u    // 2048
#define OFF_BYV   288896u    // 2048
#define OFF_BX    290944u
#define OFF_AY    291008u
#define OFF_AZ    291072u
#define OFF_BZ    291136u
#define OFF_PXM   291200u    // 2048x64
#define OFF_PYM   422272u    // 64x2048
#define OFF_PZM   553344u    // 64x64
#define OFF_Z1    557440u
#define OFF_Z2    557504u
#define OFF_PZR   557568u
#define OFF_PZC   557632u
#define OFF_PZS   557696u    // 64x64
#define OFF_M1    561792u    // 64x2048
#define OFF_G1    692864u    // 64x128
#define OFF_G2    701056u    // 64x128
#define OFF_TMP   709248u    // 128x128
#define OFF_SCAL  725632u    // [0]=diff2 [1]=normP2 [2]=cost
#define OFF_FLAGS 725636u    // int[2]: done, do_anchor
#define OFF_P     786432u    // 2048x2048

// D = A(16x4) x B(4x16) + C, fp32 throughout.
__device__ __forceinline__ v8f wmma4(v2f a, v2f b, v8f c) {
  return __builtin_amdgcn_wmma_f32_16x16x4_f32(false, a, false, b, (short)0, c,
                                               false, false);
}

// ======================= init =======================
__global__ void lot_init(float* P, float* Cx, float* Cy,
                         const float* Cx0, const float* Cy0, int* flags) {
  int idx = blockIdx.x * blockDim.x + threadIdx.x;
  P[idx] = 1.0f / ((float)N_PTS * (float)M_PTS);
  if (idx < KA * DIM) { Cx[idx] = Cx0[idx]; Cy[idx] = Cy0[idx]; }
  if (idx == 0) { flags[0] = 0; flags[1] = 0; }
}

__global__ void lot_zero(float* p, int cnt) {
  if (threadIdx.x == 0) for (int i = 0; i < cnt; ++i) p[i] = 0.0f;
}

// squared row norms: x is rows x DIM, one block (128 threads) per row
__global__ void lot_rownorm(const float* x, float* out) {
  __shared__ float sb[DIM];
  int row = blockIdx.x, tid = threadIdx.x;
  float v = x[(size_t)row * DIM + tid];
  sb[tid] = v * v;
  __syncthreads();
  for (int s = DIM / 2; s > 0; s >>= 1) {
    if (tid < s) sb[tid] += sb[tid + s];
    __syncthreads();
  }
  if (tid == 0) out[row] = sb[0];
}

// ======================= WMMA GEMM kernels =======================
// K-matrix: out[i,j] = exp(-max(an[i]+bn[j]-2*dot(A_i,B_j),0) * inv_eps)
// A: MxD row-major, B: NxD row-major (NT).
__global__ void lot_gemm_nt_exp(const float* __restrict__ A,
                                const float* __restrict__ B,
                                const float* __restrict__ an,
                                const float* __restrict__ bn,
                                float* __restrict__ out,
                                int M, int N, int D, float inv_eps) {
  int wid = blockIdx.x * WPB + (threadIdx.x >> 5);
  int lane = threadIdx.x & 31;
  int tilesN = N >> 4;
  int ti = wid / tilesN, tj = wid - ti * tilesN;
  int r0 = ti << 4, c0 = tj << 4;
  int half = lane >> 4, lr = lane & 15;
  const float* ar = A + (size_t)(r0 + lr) * D + 2 * half;
  const float* br = B + (size_t)(c0 + lr) * D + 2 * half;
  v8f acc = {0, 0, 0, 0, 0, 0, 0, 0};
  for (int kk = 0; kk < D; kk += 4) {
    v2f a, b;
    a.x = ar[kk]; a.y = ar[kk + 1];
    b.x = br[kk]; b.y = br[kk + 1];
    acc = wmma4(a, b, acc);
  }
  int col = c0 + lr;
  float bnv = bn[col];
#pragma unroll
  for (int r = 0; r < 8; ++r) {
    int row = r0 + r + 8 * half;
    float c = fmaxf(an[row] + bnv - 2.0f * acc[r], 0.0f);
    out[(size_t)row * N + col] = expf(-c * inv_eps);
  }
}

// C = A(MxK) @ B(KxN), all row-major (NN), plain store.
__global__ void lot_gemm_nn(const float* __restrict__ A,
                            const float* __restrict__ B,
                            float* __restrict__ C, int M, int N, int K) {
  int wid = blockIdx.x * WPB + (threadIdx.x >> 5);
  int lane = threadIdx.x & 31;
  int tilesN = N >> 4;
  int ti = wid / tilesN, tj = wid - ti * tilesN;
  int r0 = ti << 4, c0 = tj << 4;
  int half = lane >> 4, lr = lane & 15;
  const float* ar = A + (size_t)(r0 + lr) * K + 2 * half;
  v8f acc = {0, 0, 0, 0, 0, 0, 0, 0};
  for (int kk = 0; kk < K; kk += 4) {
    v2f a, b;
    a.x = ar[kk]; a.y = ar[kk + 1];
    b.x = B[(size_t)(kk + 2 * half) * N + c0 + lr];
    b.y = B[(size_t)(kk + 2 * half + 1) * N + c0 + lr];
    acc = wmma4(a, b, acc);
  }
  int col = c0 + lr;
#pragma unroll
  for (int r = 0; r < 8; ++r)
    C[(size_t)(r0 + r + 8 * half) * N + col] = acc[r];
}

// C = A^T @ B : A is KxM row-major, B is KxN row-major (TN).
__global__ void lot_gemm_tn(const float* __restrict__ A,
                            const float* __restrict__ B,
                            float* __restrict__ C, int M, int N, int K) {
  int wid = blockIdx.x * WPB + (threadIdx.x >> 5);
  int lane = threadIdx.x & 31;
  int tilesN = N >> 4;
  int ti = wid / tilesN, tj = wid - ti * tilesN;
  int r0 = ti << 4, c0 = tj << 4;
  int half = lane >> 4, lr = lane & 15;
  v8f acc = {0, 0, 0, 0, 0, 0, 0, 0};
  for (int kk = 0; kk < K; kk += 4) {
    v2f a, b;
    a.x = A[(size_t)(kk + 2 * half) * M + r0 + lr];
    a.y = A[(size_t)(kk + 2 * half + 1) * M + r0 + lr];
    b.x = B[(size_t)(kk + 2 * half) * N + c0 + lr];
    b.y = B[(size_t)(kk + 2 * half + 1) * N + c0 + lr];
    acc = wmma4(a, b, acc);
  }
  int col = c0 + lr;
#pragma unroll
  for (int r = 0; r < 8; ++r)
    C[(size_t)(r0 + r + 8 * half) * N + col] = acc[r];
}

// P_n = Px @ M1 (K=64); accumulate ||Pn-Pold||^2 and ||Pold||^2; store if !done.
__global__ void lot_gemm_p(const float* __restrict__ Px,
                           const float* __restrict__ M1,
                           float* __restrict__ P, float* scal,
                           const int* flags) {
  __shared__ float sb[128];
  int wid = blockIdx.x * WPB + (threadIdx.x >> 5);
  int lane = threadIdx.x & 31, tid = threadIdx.x;
  int tilesN = M_PTS >> 4;
  int ti = wid / tilesN, tj = wid - ti * tilesN;
  int r0 = ti << 4, c0 = tj << 4;
  int half = lane >> 4, lr = lane & 15;
  const float* ar = Px + (size_t)(r0 + lr) * KA + 2 * half;
  v8f acc = {0, 0, 0, 0, 0, 0, 0, 0};
  for (int kk = 0; kk < KA; kk += 4) {
    v2f a, b;
    a.x = ar[kk]; a.y = ar[kk + 1];
    b.x = M1[(size_t)(kk + 2 * half) * M_PTS + c0 + lr];
    b.y = M1[(size_t)(kk + 2 * half + 1) * M_PTS + c0 + lr];
    acc = wmma4(a, b, acc);
  }
  int done = flags[0];
  int col = c0 + lr;
  float dsum = 0.0f, psum = 0.0f;
#pragma unroll
  for (int r = 0; r < 8; ++r) {
    size_t ofs = (size_t)(r0 + r + 8 * half) * M_PTS + col;
    float pn = acc[r];
    float po = P[ofs];
    float d = pn - po;
    dsum += d * d;
    psum += po * po;
    if (!done) P[ofs] = pn;
  }
  sb[tid] = dsum; __syncthreads();
  for (int s = 64; s > 0; s >>= 1) { if (tid < s) sb[tid] += sb[tid + s]; __syncthreads(); }
  if (tid == 0) atomicAdd(&scal[0], sb[0]);
  __syncthreads();
  sb[tid] = psum; __syncthreads();
  for (int s = 64; s > 0; s >>= 1) { if (tid < s) sb[tid] += sb[tid + s]; __syncthreads(); }
  if (tid == 0) atomicAdd(&scal[1], sb[0]);
}

// cost += sum( max(sn+tn-2*src.tgt^T,0) * P ), NT over K=128
__global__ void lot_cost(const float* __restrict__ src,
                         const float* __restrict__ tgt,
                         const float* __restrict__ sn,
                         const float* __restrict__ tn,
                         const float* __restrict__ P, float* cost) {
  __shared__ float sb[128];
  int wid = blockIdx.x * WPB + (threadIdx.x >> 5);
  int lane = threadIdx.x & 31, tid = threadIdx.x;
  int tilesN = M_PTS >> 4;
  int ti = wid / tilesN, tj = wid - ti * tilesN;
  int r0 = ti << 4, c0 = tj << 4;
  int half = lane >> 4, lr = lane & 15;
  const float* ar = src + (size_t)(r0 + lr) * DIM + 2 * half;
  const float* br = tgt + (size_t)(c0 + lr) * DIM + 2 * half;
  v8f acc = {0, 0, 0, 0, 0, 0, 0, 0};
  for (int kk = 0; kk < DIM; kk += 4) {
    v2f a, b;
    a.x = ar[kk]; a.y = ar[kk + 1];
    b.x = br[kk]; b.y = br[kk + 1];
    acc = wmma4(a, b, acc);
  }
  int col = c0 + lr;
  float tnv = tn[col];
  float local = 0.0f;
#pragma unroll
  for (int r = 0; r < 8; ++r) {
    int row = r0 + r + 8 * half;
    float c = fmaxf(sn[row] + tnv - 2.0f * acc[r], 0.0f);
    local += c * P[(size_t)row * M_PTS + col];
  }
  sb[tid] = local; __syncthreads();
  for (int s = 64; s > 0; s >>= 1) { if (tid < s) sb[tid] += sb[tid + s]; __syncthreads(); }
  if (tid == 0) atomicAdd(cost, sb[0]);
}

// ======================= Sinkhorn (persistent, 1 WG x 1024 thr) =======================
__device__ __forceinline__ float safe_ratio(float num, float den) {
  return expf(fminf(logf(fmaxf(num, 0.0f)) - logf(fmaxf(den, 0.0f)), CLIPV));
}

__global__ void __launch_bounds__(1024)
lot_sinkhorn(const float* __restrict__ Kx, const float* __restrict__ Kz,
             const float* __restrict__ Ky, float* g_ax, float* g_bx,
             float* g_ay, float* g_by, float* g_az, float* g_bz) {
  __shared__ float s_ax[N_PTS], s_by[M_PTS];
  __shared__ float s_bx[KA], s_ay[KB], s_az[KA], s_bz[KB];
  __shared__ float red[1024];
  int tid = threadIdx.x;
  const float MU = 1.0f / (float)N_PTS, NU = 1.0f / (float)M_PTS;
  for (int i = tid; i < N_PTS; i += 1024) { s_ax[i] = 1.0f; s_by[i] = 1.0f; }
  if (tid < KA) { s_bx[tid] = 1.0f; s_ay[tid] = 1.0f; s_az[tid] = 1.0f; s_bz[tid] = 1.0f; }
  __syncthreads();

  for (int it = 0; it < T_NINNER; ++it) {
    // ax = ratio(mu, Kx@bx)   (2 rows / thread)
    for (int r = tid; r < N_PTS; r += 1024) {
      float s = 0.0f;
      const float* kr = Kx + (size_t)r * KA;
#pragma unroll 8
      for (int j = 0; j < KA; ++j) s += kr[j] * s_bx[j];
      s_ax[r] = safe_ratio(MU, s);
    }
    // by = ratio(nu, Ky^T@ay)  (columns, coalesced)
    for (int c = tid; c < M_PTS; c += 1024) {
      float s = 0.0f;
#pragma unroll 8
      for (int i = 0; i < KB; ++i) s += Ky[(size_t)i * M_PTS + c] * s_ay[i];
      s_by[c] = safe_ratio(NU, s);
    }
    __syncthreads();
    // kxtax[col] = sum_r Kx[r,col]*ax[r] : 16 partials per column
    {
      int col = tid & 63, sub = tid >> 6;  // 16 subs x 64 cols
      float s = 0.0f;
      for (int r = sub; r < N_PTS; r += 16) s += Kx[(size_t)r * KA + col] * s_ax[r];
      red[sub * 64 + col] = s;
    }
    __syncthreads();
    if (tid < KA) {
      float kxtax = 0.0f;
#pragma unroll
      for (int u = 0; u < 16; ++u) kxtax += red[u * 64 + tid];
      float kzbz = 0.0f;
      const float* zr = Kz + (size_t)tid * KB;
#pragma unroll 8
      for (int j = 0; j < KB; ++j) kzbz += zr[j] * s_bz[j];
      float w = sqrtf(s_az[tid] * kzbz * s_bx[tid] * kxtax);
      s_bx[tid] = safe_ratio(w, kxtax);
      s_az[tid] = safe_ratio(w, kzbz);
    }
    __syncthreads();
    // kyby[row] = sum_c Ky[row,c]*by[c] : 16 partials per row
    {
      int row = tid >> 4, sub = tid & 15;
      float s = 0.0f;
      for (int c = sub; c < M_PTS; c += 16) s += Ky[(size_t)row * M_PTS + c] * s_by[c];
      red[tid] = s;
    }
    __syncthreads();
    if (tid < KB) {
      float kyby = 0.0f;
#pragma unroll
      for (int u = 0; u < 16; ++u) kyby += red[tid * 16 + u];
      float kztaz = 0.0f;
#pragma unroll 8
      for (int j = 0; j < KA; ++j) kztaz += Kz[(size_t)j * KB + tid] * s_az[j];
      float w = sqrtf(s_ay[tid] * kyby * s_bz[tid] * kztaz);
      s_bz[tid] = safe_ratio(w, kztaz);
      s_ay[tid] = safe_ratio(w, kyby);
    }
    __syncthreads();
  }
  for (int i = tid; i < N_PTS; i += 1024) { g_ax[i] = s_ax[i]; g_by[i] = s_by[i]; }
  if (tid < KA) { g_bx[tid] = s_bx[tid]; g_ay[tid] = s_ay[tid];
                  g_az[tid] = s_az[tid]; g_bz[tid] = s_bz[tid]; }
}

// ======================= elementwise / reductions =======================
__global__ void lot_scale(float* __restrict__ out, const float* __restrict__ Kmat,
                          const float* __restrict__ a, const float* __restrict__ b,
                          int R, int C) {
  int idx = blockIdx.x * blockDim.x + threadIdx.x;
  int i = idx / C, j = idx - i * C;
  out[idx] = a[i] * Kmat[idx] * b[j];
}

__global__ void lot_colsum(const float* __restrict__ A, float* z, int R, int C) {
  __shared__ float sb[256];
  int j = blockIdx.x, tid = threadIdx.x;
  float s = 0.0f;
  for (int i = tid; i < R; i += 256) s += A[(size_t)i * C + j];
  sb[tid] = s; __syncthreads();
  for (int t = 128; t > 0; t >>= 1) { if (tid < t) sb[tid] += sb[tid + t]; __syncthreads(); }
  if (tid == 0) z[j] = sb[0];
}

__global__ void lot_rowsum(const float* __restrict__ A, float* z, int C) {
  __shared__ float sb[256];
  int i = blockIdx.x, tid = threadIdx.x;
  float s = 0.0f;
  for (int j = tid; j < C; j += 256) s += A[(size_t)i * C + j];
  sb[tid] = s; __syncthreads();
  for (int t = 128; t > 0; t >>= 1) { if (tid < t) sb[tid] += sb[tid + t]; __syncthreads(); }
  if (tid == 0) z[i] = sb[0];
}

__global__ void lot_pzsums(const float* __restrict__ Pz, float* pzr, float* pzc) {
  int t = threadIdx.x;
  float r = 0.0f, c = 0.0f;
  for (int j = 0; j < KB; ++j) r += Pz[t * KB + j];
  for (int i = 0; i < KA; ++i) c += Pz[i * KB + t];
  pzr[t] = r; pzc[t] = c;
}

__global__ void lot_pzs(const float* __restrict__ Pz, const float* __restrict__ z1,
                        const float* __restrict__ z2, float* __restrict__ Pzs) {
  int idx = blockIdx.x * blockDim.x + threadIdx.x;
  int l = idx >> 6, t = idx & 63;
  Pzs[idx] = Pz[idx] / (z1[l] * z2[t]);
}

__global__ void lot_temp_build(const float* __restrict__ Pz, const float* __restrict__ z1,
                               const float* __restrict__ z2, const float* __restrict__ pzr,
                               const float* __restrict__ pzc, float* __restrict__ temp) {
  int idx = blockIdx.x * blockDim.x + threadIdx.x;  // 128x128
  int r = idx >> 7, c = idx & 127;
  float v;
  if (r < KA && c < KA)       v = (r == c) ? (z1[r] + pzr[r]) : 0.0f;
  else if (r < KA)            v = -Pz[r * KB + (c - KA)];
  else if (c < KA)            v = -Pz[c * KB + (r - KA)];
  else                        v = (r == c) ? (z2[r - KA] + pzc[r - KA]) : 0.0f;
  temp[idx] = v;
}

__global__ void lot_done_update(float* scal, int* flags, int t) {
  float converr = sqrtf(scal[0]) / sqrtf(scal[1]);
  int nd = (flags[0] != 0) || (converr < TOLRATIO);
  flags[1] = (!nd && (t < T_NITER - 1)) ? 1 : 0;
  flags[0] = nd;
}

// Gauss-Jordan solve: temp (128x128, symmetric diag-dominant) * Y = rhs^T,
// rhs^T rows 0..63 = G1, rows 64..127 = G2.  Cx = Y[0:64], Cy = Y[64:128].
__global__ void lot_solve(const float* __restrict__ temp, const float* __restrict__ G1,
                          const float* __restrict__ G2, float* Cx, float* Cy,
                          const int* flags) {
  if (!flags[1]) return;  // do_anchor gate (uniform)
  extern __shared__ float sm[];
  float* T = sm;           // 128*128
  float* R = sm + 16384;   // 128*128
  __shared__ float fac[128];
  int tid = threadIdx.x;   // 256 threads
  for (int i = tid; i < 16384; i += 256) {
    T[i] = temp[i];
    int r = i >> 7, c = i & 127;
    R[i] = (r < KA) ? G1[r * DIM + c] : G2[(r - KA) * DIM + c];
  }
  __syncthreads();
  for (int k = 0; k < 128; ++k) {
    float piv = 1.0f / T[k * 128 + k];
    __syncthreads();
    for (int c = tid; c < 256; c += 256) {
      if (c < 128) T[k * 128 + c] *= piv;
      else         R[k * 128 + (c - 128)] *= piv;
    }
    __syncthreads();
    for (int i = tid; i < 128; i += 256) fac[i] = T[i * 128 + k];
    __syncthreads();
    for (int idx = tid; idx < 128 * 256; idx += 256) {
      int i = idx >> 8, c = idx & 255;
      if (i == k) continue;
      if (c < 128) T[i * 128 + c]         -= fac[i] * T[k * 128 + c];
      else         R[i * 128 + (c - 128)] -= fac[i] * R[k * 128 + (c - 128)];
    }
    __syncthreads();
  }
  for (int i = tid; i < 16384; i += 256) {
    int r = i >> 7, c = i & 127;
    if (r < KA) Cx[r * DIM + c] = R[i];
    else        Cy[(r - KA) * DIM + c] = R[i];
  }
}

__global__ void lot_writeout(const float* scal, const float* Cx, const float* Cy,
                             float* out) {
  int idx = blockIdx.x * blockDim.x + threadIdx.x;
  if (idx == 0) out[0] = scal[2];
  else if (idx <= KA * DIM) out[idx] = Cx[idx - 1];
  else if (idx <= 2 * KA * DIM) out[idx] = Cy[idx - 1 - KA * DIM];
}

// ======================= host orchestration =======================
extern "C" void kernel_launch(void* const* d_in, const int* in_sizes, int n_in,
                              void* d_out, int out_size, void* d_ws, size_t ws_size,
                              hipStream_t stream) {
  (void)in_sizes; (void)n_in; (void)out_size; (void)ws_size;
  const float* src = (const float*)d_in[0];
  const float* tgt = (const float*)d_in[1];
  const float* Cx0 = (const float*)d_in[2];
  const float* Cy0 = (const float*)d_in[3];
  float* w = (float*)d_ws;
  float *Cx = w + OFF_CX, *Cy = w + OFF_CY;
  float *sn = w + OFF_SN, *tn = w + OFF_TN, *cxn = w + OFF_CXN, *cyn = w + OFF_CYN;
  float *Kx = w + OFF_KX, *Kz = w + OFF_KZ, *Ky = w + OFF_KY;
  float *ax = w + OFF_AX, *byv = w + OFF_BYV, *bx = w + OFF_BX;
  float *ay = w + OFF_AY, *az = w + OFF_AZ, *bz = w + OFF_BZ;
  float *Px = w + OFF_PXM, *Py = w + OFF_PYM, *Pz = w + OFF_PZM;
  float *z1 = w + OFF_Z1, *z2 = w + OFF_Z2, *pzr = w + OFF_PZR, *pzc = w + OFF_PZC;
  float *Pzs = w + OFF_PZS, *M1 = w + OFF_M1, *G1 = w + OFF_G1, *G2 = w + OFF_G2;
  float *temp = w + OFF_TMP, *scal = w + OFF_SCAL, *P = w + OFF_P;
  int* flags = (int*)(w + OFF_FLAGS);

  lot_init<<<(N_PTS * M_PTS) / 256, 256, 0, stream>>>(P, Cx, Cy, Cx0, Cy0, flags);
  lot_rownorm<<<N_PTS, DIM, 0, stream>>>(src, sn);
  lot_rownorm<<<M_PTS, DIM, 0, stream>>>(tgt, tn);

  for (int t = 0; t < T_NITER; ++t) {
    lot_rownorm<<<KA, DIM, 0, stream>>>(Cx, cxn);
    lot_rownorm<<<KB, DIM, 0, stream>>>(Cy, cyn);
    // kernel matrices (WMMA, exp epilogue)
    lot_gemm_nt_exp<<<(N_PTS / 16) * (KA / 16) / WPB, 128, 0, stream>>>(
        src, Cx, sn, cxn, Kx, N_PTS, KA, DIM, 1.0f / 64.0f);
    lot_gemm_nt_exp<<<(KA / 16) * (KB / 16) / WPB, 128, 0, stream>>>(
        Cx, Cy, cxn, cyn, Kz, KA, KB, DIM, 1.0f / 64.0f);
    lot_gemm_nt_exp<<<(KB / 16) * (M_PTS / 16) / WPB, 128, 0, stream>>>(
        Cy, tgt, cyn, tn, Ky, KB, M_PTS, DIM, 1.0f / 64.0f);
    // 100-step coupled Sinkhorn, fused into one persistent workgroup
    lot_sinkhorn<<<1, 1024, 0, stream>>>(Kx, Kz, Ky, ax, bx, ay, byv, az, bz);
    // transport plans
    lot_scale<<<(N_PTS * KA) / 256, 256, 0, stream>>>(Px, Kx, ax, bx, N_PTS, KA);
    lot_scale<<<(KA * KB) / 256, 256, 0, stream>>>(Pz, Kz, az, bz, KA, KB);
    lot_scale<<<(KB * M_PTS) / 256, 256, 0, stream>>>(Py, Ky, ay, byv, KB, M_PTS);
    lot_colsum<<<KA, 256, 0, stream>>>(Px, z1, N_PTS, KA);
    lot_rowsum<<<KB, 256, 0, stream>>>(Py, z2, M_PTS);
    lot_pzsums<<<1, 64, 0, stream>>>(Pz, pzr, pzc);
    lot_pzs<<<(KA * KB) / 256, 256, 0, stream>>>(Pz, z1, z2, Pzs);
    // M1 = (Pz/(z1 z2)) @ Py  -> 64x2048    (WMMA)
    lot_gemm_nn<<<(KA / 16) * (M_PTS / 16) / WPB, 128, 0, stream>>>(
        Pzs, Py, M1, KA, M_PTS, KB);
    // P_n = Px @ M1 -> 2048x2048; fused converr norms; store gated on old done
    lot_zero<<<1, 1, 0, stream>>>(scal, 2);
    lot_gemm_p<<<(N_PTS / 16) * (M_PTS / 16) / WPB, 128, 0, stream>>>(
        Px, M1, P, scal, flags);
    lot_done_update<<<1, 1, 0, stream>>>(scal, flags, t);
    // anchor update: rhs^T rows = [Px^T@src ; Py@tgt]  (WMMA)
    lot_gemm_tn<<<(KA / 16) * (DIM / 16) / WPB, 128, 0, stream>>>(
        Px, src, G1, KA, DIM, N_PTS);
    lot_gemm_nn<<<(KB / 16) * (DIM / 16) / WPB, 128, 0, stream>>>(
        Py, tgt, G2, KB, DIM, M_PTS);
    lot_temp_build<<<(128 * 128) / 256, 256, 0, stream>>>(Pz, z1, z2, pzr, pzc, temp);
    lot_solve<<<1, 256, 131072, stream>>>(temp, G1, G2, Cx, Cy, flags);
  }
  // cost = sum(P * C) with C recomputed on the fly (WMMA + reduce)
  lot_zero<<<1, 1, 0, stream>>>(scal + 2, 1);
  lot_cost<<<(N_PTS / 16) * (M_PTS / 16) / WPB, 128, 0, stream>>>(
      src, tgt, sn, tn, P, scal + 2);
  lot_writeout<<<(1 + 2 * KA * DIM + 255) / 256, 256, 0, stream>>>(
      scal, Cx, Cy, (float*)d_out);
}